// CTC_CRF_Torch_61306363183179
// MI455X (gfx1250) — compile-verified
//
#include <hip/hip_runtime.h>

#define T_STEPS 800
#define BATCH   32
#define NSTATE  1024
#define NZ      5
#define ROW     (NSTATE * NZ)   // 5120 floats per (t, n)
#define ROW4    (ROW / 4)       // 1280 float4
#define THREADS 256
#define DEPTH   4               // prefetch distance (steps)
#define NBUF    6               // LDS ring slots = DEPTH + 2 (single-barrier scheme)

typedef __attribute__((ext_vector_type(4))) float v4f;

// ---------------------------------------------------------------------------
// CDNA5 async global->LDS copy (ASYNCcnt-tracked) + wait
// ---------------------------------------------------------------------------
#if defined(__HIP_DEVICE_COMPILE__)
#if __has_builtin(__builtin_amdgcn_s_wait_asynccnt)
#define WAIT_ASYNC(n) __builtin_amdgcn_s_wait_asynccnt(n)
#else
#define WAIT_ASYNC(n) asm volatile("s_wait_asynccnt " #n ::: "memory")
#endif
#else
#define WAIT_ASYNC(n)
#endif

__device__ __forceinline__ void async_ld16(void* lds_dst, const void* gsrc) {
#if defined(__HIP_DEVICE_COMPILE__)
  // Low 32 bits of a generic pointer into LDS are the wave-relative LDS offset.
  unsigned l = (unsigned)(unsigned long long)lds_dst;
  asm volatile("global_load_async_to_lds_b128 %0, %1, off"
               :
               : "v"(l), "v"(gsrc)
               : "memory");
#endif
}

__device__ __forceinline__ void store_nt4(v4f* p, v4f v) {
#if __has_builtin(__builtin_nontemporal_store)
  __builtin_nontemporal_store(v, p);
#else
  *p = v;
#endif
}

__device__ __forceinline__ float lse5(float t0, float t1, float t2, float t3, float t4) {
  float mx = fmaxf(fmaxf(fmaxf(t0, t1), fmaxf(t2, t3)), t4);
  return mx + __logf(__expf(t0 - mx) + __expf(t1 - mx) + __expf(t2 - mx) +
                     __expf(t3 - mx) + __expf(t4 - mx));
}

// ---------------------------------------------------------------------------
// Fused forward/backward scan. grid = 2*BATCH blocks: block = (n, dir).
// Forward blocks also emit the transposed score copy (row already in LDS,
// non-temporal store) and stage fwd alphas into the posts region.
// Pipeline (per step): issue batch t+DEPTH -> s_wait_asynccnt 5*DEPTH
// (in-order => batch t landed) -> ONE barrier -> compute.
// ---------------------------------------------------------------------------
__global__ __launch_bounds__(THREADS)
void fb_scan_kernel(const float* __restrict__ scores,   // (T, N, ROW)
                    float* __restrict__ outT,           // (N, T, ROW)
                    float* __restrict__ bwd,            // (N, T+1, S)
                    float* __restrict__ fwd)            // (N, T+1, S)  (posts region)
{
  __shared__ float4 sbuf[NBUF][ROW4];       // 6-slot score ring (120 KB)
  __shared__ float4 spub[2][NSTATE / 4];    // double-buffered published state (8 KB)

  const int tid = threadIdx.x;
  const int n   = blockIdx.x >> 1;
  const int dir = blockIdx.x & 1;

  const float4 z4 = make_float4(0.f, 0.f, 0.f, 0.f);
  spub[0][tid] = z4;                         // alpha/beta at boundary = 0
  float a0 = 0.f, a1 = 0.f, a2 = 0.f, a3 = 0.f;   // this thread's 4 states

  if (dir == 0) {
    // ============================ FORWARD ============================
    float* dst = fwd + (size_t)n * (T_STEPS + 1) * NSTATE;
    ((float4*)dst)[tid] = z4;                // fwd[0] = v0 = 0

    // prologue: prefetch batches t = 0..DEPTH-1 into slots 0..DEPTH-1
    for (int t = 0; t < DEPTH; ++t) {
      const float4* g = (const float4*)(scores + ((size_t)t * BATCH + n) * ROW);
      #pragma unroll
      for (int j = 0; j < 5; ++j)
        async_ld16(&sbuf[t][tid + j * THREADS], g + tid + j * THREADS);
    }

    int cur = 0;
    int is = DEPTH;                          // slot for batch t+DEPTH
    int cs = 0;                              // slot for batch t
    for (int t = 0; t < T_STEPS; ++t) {
      if (t + DEPTH < T_STEPS) {
        const float4* g = (const float4*)(scores + ((size_t)(t + DEPTH) * BATCH + n) * ROW);
        float4* lb = sbuf[is];
        #pragma unroll
        for (int j = 0; j < 5; ++j)
          async_ld16(&lb[tid + j * THREADS], g + tid + j * THREADS);
        WAIT_ASYNC(20);                      // 5*DEPTH newer => batch t complete
      } else {
        WAIT_ASYNC(0);
      }
      __syncthreads();                       // batch t visible + state t-1 published

      const float4* sc4 = sbuf[cs];
      const float*  stc = (const float*)spub[cur];
      // cross-state gather: idx[s][1+j] = j*256 + s/4 ; s/4 == tid for s in [4tid,4tid+3]
      float b0 = stc[tid], b1 = stc[256 + tid], b2 = stc[512 + tid], b3 = stc[768 + tid];

      float f[20];
      #pragma unroll
      for (int q = 0; q < 5; ++q) ((float4*)f)[q] = sc4[5 * tid + q];

      float r0 = lse5(f[0]  + a0, f[1]  + b0, f[2]  + b1, f[3]  + b2, f[4]  + b3);
      float r1 = lse5(f[5]  + a1, f[6]  + b0, f[7]  + b1, f[8]  + b2, f[9]  + b3);
      float r2 = lse5(f[10] + a2, f[11] + b0, f[12] + b1, f[13] + b2, f[14] + b3);
      float r3 = lse5(f[15] + a3, f[16] + b0, f[17] + b1, f[18] + b2, f[19] + b3);
      a0 = r0; a1 = r1; a2 = r2; a3 = r3;

      float4 r4 = make_float4(r0, r1, r2, r3);
      spub[cur ^ 1][tid] = r4;
      ((float4*)(dst + (size_t)(t + 1) * NSTATE))[tid] = r4;

      // fused transposed score copy (write-once stream -> non-temporal)
      const v4f* scv = (const v4f*)sc4;
      v4f* trow = (v4f*)(outT + ((size_t)n * T_STEPS + t) * ROW);
      #pragma unroll
      for (int j = 0; j < 5; ++j)
        store_nt4(&trow[tid + j * THREADS], scv[tid + j * THREADS]);

      cur ^= 1;
      is = (is + 1 == NBUF) ? 0 : is + 1;
      cs = (cs + 1 == NBUF) ? 0 : cs + 1;
    }
  } else {
    // ============================ BACKWARD ===========================
    float* dst = bwd + (size_t)n * (T_STEPS + 1) * NSTATE;
    ((float4*)(dst + (size_t)T_STEPS * NSTATE))[tid] = z4;   // bwd[T] = v0 = 0

    for (int u = 0; u < DEPTH; ++u) {        // prefetch tt = T-1 .. T-DEPTH
      const float4* g = (const float4*)(scores + ((size_t)(T_STEPS - 1 - u) * BATCH + n) * ROW);
      #pragma unroll
      for (int j = 0; j < 5; ++j)
        async_ld16(&sbuf[u][tid + j * THREADS], g + tid + j * THREADS);
    }

    // For state v: beta[t][v] = LSE( M[t][5v] + beta[t+1][v],
    //    { M[t][20*(v%256)+5r+1+(v>>8)] + beta[t+1][4*(v%256)+r] : r=0..3 } )
    const int lob = tid & 63;                // (4*tid % 256)/4
    const int hi  = tid >> 6;                // v >> 8 (uniform over this thread's 4 states)
    const int eb0 = 80 * lob + 1 + hi;

    int cur = 0;
    int is = DEPTH;
    int cs = 0;
    for (int u = 0; u < T_STEPS; ++u) {
      const int tt = T_STEPS - 1 - u;
      if (u + DEPTH < T_STEPS) {
        const float4* g = (const float4*)(scores + ((size_t)(tt - DEPTH) * BATCH + n) * ROW);
        float4* lb = sbuf[is];
        #pragma unroll
        for (int j = 0; j < 5; ++j)
          async_ld16(&lb[tid + j * THREADS], g + tid + j * THREADS);
        WAIT_ASYNC(20);
      } else {
        WAIT_ASYNC(0);
      }
      __syncthreads();

      const float*  sc  = (const float*)sbuf[cs];
      const float4* bc4 = spub[cur];
      float4 c0 = bc4[4 * lob + 0];
      float4 c1 = bc4[4 * lob + 1];
      float4 c2 = bc4[4 * lob + 2];
      float4 c3 = bc4[4 * lob + 3];
      const float* sm = sc + 20 * tid;       // self scores: 5v = 20*tid + 5k

      float r0 = lse5(sm[0]  + a0, sc[eb0]      + c0.x, sc[eb0 + 5]  + c0.y,
                      sc[eb0 + 10] + c0.z, sc[eb0 + 15] + c0.w);
      float r1 = lse5(sm[5]  + a1, sc[eb0 + 20] + c1.x, sc[eb0 + 25] + c1.y,
                      sc[eb0 + 30] + c1.z, sc[eb0 + 35] + c1.w);
      float r2 = lse5(sm[10] + a2, sc[eb0 + 40] + c2.x, sc[eb0 + 45] + c2.y,
                      sc[eb0 + 50] + c2.z, sc[eb0 + 55] + c2.w);
      float r3 = lse5(sm[15] + a3, sc[eb0 + 60] + c3.x, sc[eb0 + 65] + c3.y,
                      sc[eb0 + 70] + c3.z, sc[eb0 + 75] + c3.w);
      a0 = r0; a1 = r1; a2 = r2; a3 = r3;

      float4 r4 = make_float4(r0, r1, r2, r3);
      spub[cur ^ 1][tid] = r4;
      ((float4*)(dst + (size_t)tt * NSTATE))[tid] = r4;

      cur ^= 1;
      is = (is + 1 == NBUF) ? 0 : is + 1;
      cs = (cs + 1 == NBUF) ? 0 : cs + 1;
    }
  }
}

// ---------------------------------------------------------------------------
// posts[n][t][:] = softmax(fwd + bwd) over S; fwd is in-place in `posts`.
// One block (8 wave32s) per (n, t) row; wave-shuffle reductions.
// ---------------------------------------------------------------------------
__global__ __launch_bounds__(THREADS)
void posts_kernel(float* __restrict__ posts, const float* __restrict__ bwd) {
  __shared__ float red[THREADS / 32];
  const int tid  = threadIdx.x;
  const int lane = tid & 31;
  const int wid  = tid >> 5;
  const size_t base = (size_t)blockIdx.x * NSTATE;

  float4 fv = ((const float4*)(posts + base))[tid];
  float4 bv = ((const float4*)(bwd + base))[tid];
  float x0 = fv.x + bv.x, x1 = fv.y + bv.y, x2 = fv.z + bv.z, x3 = fv.w + bv.w;

  float mx = fmaxf(fmaxf(x0, x1), fmaxf(x2, x3));
  #pragma unroll
  for (int m = 16; m > 0; m >>= 1) mx = fmaxf(mx, __shfl_xor(mx, m, 32));
  if (lane == 0) red[wid] = mx;
  __syncthreads();
  mx = red[0];
  #pragma unroll
  for (int w = 1; w < THREADS / 32; ++w) mx = fmaxf(mx, red[w]);
  __syncthreads();                                  // red about to be reused

  float e0 = __expf(x0 - mx), e1 = __expf(x1 - mx), e2 = __expf(x2 - mx), e3 = __expf(x3 - mx);
  float sum = e0 + e1 + e2 + e3;
  #pragma unroll
  for (int m = 16; m > 0; m >>= 1) sum += __shfl_xor(sum, m, 32);
  if (lane == 0) red[wid] = sum;
  __syncthreads();
  sum = red[0];
  #pragma unroll
  for (int w = 1; w < THREADS / 32; ++w) sum += red[w];

  float inv = 1.0f / sum;
  ((float4*)(posts + base))[tid] = make_float4(e0 * inv, e1 * inv, e2 * inv, e3 * inv);
}

// ---------------------------------------------------------------------------
extern "C" void kernel_launch(void* const* d_in, const int* in_sizes, int n_in,
                              void* d_out, int out_size, void* d_ws, size_t ws_size,
                              hipStream_t stream) {
  (void)in_sizes; (void)n_in; (void)out_size; (void)d_ws; (void)ws_size;
  const float* scores = (const float*)d_in[0];
  float* out   = (float*)d_out;
  float* outT  = out;                                              // (N, T, 5120)
  float* bwd   = outT + (size_t)BATCH * T_STEPS * ROW;             // (N, T+1, S)
  float* posts = bwd + (size_t)BATCH * (T_STEPS + 1) * NSTATE;     // (N, T+1, S)

  fb_scan_kernel<<<2 * BATCH, THREADS, 0, stream>>>(scores, outT, bwd, posts);
  posts_kernel<<<BATCH * (T_STEPS + 1), THREADS, 0, stream>>>(posts, bwd);
}